// feature_aggregation_10161892622587
// MI455X (gfx1250) — compile-verified
//
#include <hip/hip_runtime.h>

// Problem sizes (fixed by the reference).
#define B_ 4
#define T_ 12
#define N_ 4096
#define K_ 8
#define D_ 64
#define BT_ (B_ * T_)               // 48
#define NODES_ (BT_ * N_)           // 196608 (b,t,n) nodes
#define OUT_ELEMS_ (NODES_ * D_)    // 12,582,912 floats
#define PAIRS_ (NODES_ / 2)         // one wave32 per node pair

typedef __attribute__((ext_vector_type(2))) float v2f;
typedef __attribute__((ext_vector_type(8))) float v8f;

__global__ __launch_bounds__(256) void fa_init_kernel(float* __restrict__ out,
                                                      float* __restrict__ counts) {
    int i = blockIdx.x * blockDim.x + threadIdx.x;
    if (i < OUT_ELEMS_) out[i] = 0.0f;
    if (i < NODES_) counts[i] = 1e-14f;   // EPS accumulator init (matches reference)
}

// One wave per node pair (n0 = 2p, n1 = 2p+1).
// A (16x16, block-diag of the two 8x8 adj) x B (16x64 gathered neighbor rows)
// = D (16x64): rows 0-7 -> n0 messages, rows 8-15 -> n1 messages.
// 4 k-chunks x 4 d-chunks of v_wmma_f32_16x16x4_f32 (16 WMMAs/wave, fp32 exact).
// Per-bt byte offsets fit in u32, so all gathers/scatters use a uniform SGPR base
// plus a zext'd 32-bit VGPR offset (GVS addressing: saddr + voffset).
__global__ __launch_bounds__(256) void fa_agg_kernel(const float* __restrict__ x,
                                                     const float* __restrict__ adj,
                                                     const int* __restrict__ indices,
                                                     float* __restrict__ out,
                                                     float* __restrict__ counts) {
    const int lane = threadIdx.x & 31;
    const int wave = (blockIdx.x * blockDim.x + threadIdx.x) >> 5;   // grid sized exactly

    const int bt = wave / (N_ / 2);
    const int p  = wave % (N_ / 2);
    const int n0 = p * 2;

    // Uniform (wave-invariant) base pointers -> SGPRs.
    const float* __restrict__ xb  = x   + (size_t)bt * (N_ * D_);
    const float* __restrict__ ab  = adj + ((size_t)bt * N_ + n0) * (K_ * K_);
    const int*   __restrict__ ib  = indices + ((size_t)bt * N_ + n0) * K_;
    float*       __restrict__ ob  = out    + (size_t)bt * (N_ * D_);
    float*       __restrict__ cb  = counts + (size_t)bt * N_;

    const int  hi     = lane >> 4;     // lane half
    const int  li     = lane & 15;     // position within half
    const int  m      = li;            // A-matrix row held by this lane (both halves)
    const bool upper  = (hi != 0);
    const bool lowBlk = (m < 8);       // this A row belongs to node n0's diagonal block

    // ---- All 16 pair indices via 4 x b128 (wave-uniform addresses). ----
    const int4* qp = (const int4*)ib;
    const int4 q0 = qp[0], q1 = qp[1], q2 = qp[2], q3 = qp[3];

    // ---- adj fragment data: two b64 loads per lane, masked with cndmask.
    // Lane needs adjpair[m*8 + {2hi, 2hi+1}] (k-chunks 0,2) and [+ {4+2hi, 5+2hi}] (1,3).
    const unsigned arow = (unsigned)(m * 8 + 2 * hi);
    const v2f a01 = *(const v2f*)(ab + arow);
    const v2f a45 = *(const v2f*)(ab + arow + 4u);

    v2f aFrag[4];
    // c=0: K columns 2hi,2hi+1   -> valid for rows m<8  (n0 block)
    aFrag[0].x = lowBlk ? a01.x : 0.0f;  aFrag[0].y = lowBlk ? a01.y : 0.0f;
    // c=1: K columns 4+2hi,5+2hi -> valid for rows m<8
    aFrag[1].x = lowBlk ? a45.x : 0.0f;  aFrag[1].y = lowBlk ? a45.y : 0.0f;
    // c=2: K columns 8+2hi,...   -> valid for rows m>=8 (n1 block)
    aFrag[2].x = lowBlk ? 0.0f : a01.x;  aFrag[2].y = lowBlk ? 0.0f : a01.y;
    // c=3: K columns 12+2hi,...  -> valid for rows m>=8
    aFrag[3].x = lowBlk ? 0.0f : a45.x;  aFrag[3].y = lowBlk ? 0.0f : a45.y;

    // B-row source indices per k-chunk: rows kk0 = 4c+2hi, kk1 = kk0+1.
    const unsigned s0[4] = { (unsigned)(upper ? q0.z : q0.x), (unsigned)(upper ? q1.z : q1.x),
                             (unsigned)(upper ? q2.z : q2.x), (unsigned)(upper ? q3.z : q3.x) };
    const unsigned s1[4] = { (unsigned)(upper ? q0.w : q0.y), (unsigned)(upper ? q1.w : q1.y),
                             (unsigned)(upper ? q2.w : q2.y), (unsigned)(upper ? q3.w : q3.y) };

    v8f acc[4] = {};   // 4 column chunks of the 16x64 f32 result

#pragma unroll
    for (int c = 0; c < 4; ++c) {
        const unsigned r0 = s0[c] * (unsigned)D_ + (unsigned)li;  // 32-bit voffset
        const unsigned r1 = s1[c] * (unsigned)D_ + (unsigned)li;
#pragma unroll
        for (int dc = 0; dc < 4; ++dc) {
            v2f b;
            b.x = xb[r0 + dc * 16];
            b.y = xb[r1 + dc * 16];
            acc[dc] = __builtin_amdgcn_wmma_f32_16x16x4_f32(
                false, aFrag[c], false, b, (short)0, acc[dc], false, false);
        }
    }

    // Scatter destinations: D VGPR r holds row r (lanes 0-15) and row r+8 (lanes 16-31).
    const unsigned dest[8] = { (unsigned)(upper ? q2.x : q0.x), (unsigned)(upper ? q2.y : q0.y),
                               (unsigned)(upper ? q2.z : q0.z), (unsigned)(upper ? q2.w : q0.w),
                               (unsigned)(upper ? q3.x : q1.x), (unsigned)(upper ? q3.y : q1.y),
                               (unsigned)(upper ? q3.z : q1.z), (unsigned)(upper ? q3.w : q1.w) };

#pragma unroll
    for (int r = 0; r < 8; ++r) {
        const unsigned obase = dest[r] * (unsigned)D_ + (unsigned)li;  // 32-bit voffset
#pragma unroll
        for (int dc = 0; dc < 4; ++dc) {
            atomicAdd(ob + obase + dc * 16, acc[dc][r]);
        }
    }

    // Message counts: 16 messages per pair, one per lane 0-15.
    if (lane < 16) {
        const unsigned dl = (unsigned)ib[lane];
        atomicAdd(cb + dl, 1.0f);
    }
}

__global__ __launch_bounds__(256) void fa_div_kernel(float* __restrict__ out,
                                                     const float* __restrict__ counts) {
    int i = blockIdx.x * blockDim.x + threadIdx.x;
    if (i < OUT_ELEMS_) out[i] /= counts[i >> 6];   // D_ == 64
}

extern "C" void kernel_launch(void* const* d_in, const int* in_sizes, int n_in,
                              void* d_out, int out_size, void* d_ws, size_t ws_size,
                              hipStream_t stream) {
    // setup_inputs order: x, adj, batch_indices, time_indices, indices
    const float* x       = (const float*)d_in[0];
    const float* adj     = (const float*)d_in[1];
    const int*   indices = (const int*)d_in[4];
    float* out    = (float*)d_out;
    float* counts = (float*)d_ws;   // NODES_ floats of scratch

    const int threads = 256;
    const int init_blocks = (OUT_ELEMS_ + threads - 1) / threads;
    const int agg_blocks  = (PAIRS_ * 32) / threads;   // 8 waves (pairs) per block

    fa_init_kernel<<<init_blocks, threads, 0, stream>>>(out, counts);
    fa_agg_kernel<<<agg_blocks, threads, 0, stream>>>(x, adj, indices, out, counts);
    fa_div_kernel<<<init_blocks, threads, 0, stream>>>(out, counts);
}